// StreamDeepVQE_37606733644447
// MI455X (gfx1250) — compile-verified
//
#include <hip/hip_runtime.h>
#include <math.h>

// ---------------------------------------------------------------------------
// StreamDeepVQE forward for MI455X (gfx1250).
// fp32 end-to-end. Convs use V_WMMA_F32_16X16X4_F32 (wave32, one 16x16 tile
// per wave) reading a zero-padded pre-packed input (C,4,Fin+2) so every
// inner-loop load is unconditional (no exec-mask branches). GRU mat-vecs use
// float4 VALU dot products spread across many workgroups.
// ---------------------------------------------------------------------------

typedef __attribute__((ext_vector_type(2))) float v2f;
typedef __attribute__((ext_vector_type(8))) float v8f;

__device__ __forceinline__ float elu1(float v) {
    return v > 0.0f ? v : (expf(v) - 1.0f);
}
__device__ __forceinline__ float sigm(float v) {
    return 1.0f / (1.0f + expf(-v));
}

// ------------------------- batched pack init -------------------------------
// For every stream conv, build inpP:(C,4,F+2): rows 0..2 = cache (zero-padded
// freq borders), row 3 = 0 (filled later by the producing kernel's epilogue).
struct PackJob { const float* cin; float* out; int C; int F; };
struct PackJobs { PackJob j[20]; };

__global__ void k_pack_caches(PackJobs jobs) {
    PackJob jb = jobs.j[blockIdx.y];
    int Wp = jb.F + 2;
    int n = jb.C * 4 * Wp;
    for (int e = blockIdx.x * blockDim.x + threadIdx.x; e < n;
         e += gridDim.x * blockDim.x) {
        int c = e / (4 * Wp);
        int rem = e - c * 4 * Wp;
        int t = rem / Wp;
        int w = rem - t * Wp;
        float v = 0.0f;
        if (t < 3 && w >= 1 && w <= jb.F)
            v = jb.cin[((size_t)c * 3 + t) * jb.F + (w - 1)];
        jb.out[e] = v;
    }
}

// ------------------------- feature extraction ------------------------------
// x:(257,1,2) -> en0:(2,257) plain + row 3 of enc1-conv pack (2,4,259)
__global__ void k_fe(const float* __restrict__ x, float* __restrict__ en0,
                     float* __restrict__ pack) {
    int f = blockIdx.x * blockDim.x + threadIdx.x;
    if (f >= 257) return;
    float re = x[2 * f], im = x[2 * f + 1];
    float mag = sqrtf(re * re + im * im + 1e-12f);
    float s = 1.0f / (powf(mag, 0.7f) + 1e-12f);
    float a = re * s, b = im * s;
    en0[f]       = a;
    en0[257 + f] = b;
    const int Wp = 259;
    pack[(0 * 4 + 3) * Wp + 1 + f] = a;
    pack[(1 * 4 + 3) * Wp + 1 + f] = b;
}

// ------------------------- stream conv (4x3) as WMMA GEMM ------------------
// inpP:(ci,4,Fin+2) packed zero-padded input; W:(co, ci*12) OIHW-flat.
// flags: bit0 = BN+ELU epilogue, bit1 = residual add from addp.
// packp (optional): next conv's packed input; we write row 3, width Fout+2.
__global__ void k_conv(const float* __restrict__ inpP,
                       const float* __restrict__ W,
                       const float* __restrict__ bias,
                       const float* __restrict__ bng, const float* __restrict__ bnb,
                       const float* __restrict__ bnm, const float* __restrict__ bnv,
                       const float* __restrict__ addp,
                       float* __restrict__ out,
                       float* __restrict__ packp,
                       int ci, int co, int Fin, int Fout, int stride, int flags) {
    const int lane = threadIdx.x;
    const int half = lane >> 4;       // 0 or 1
    const int l15  = lane & 15;
    const int koff = half * 2;        // K sub-offset per WMMA f32 A/B layout
    const int co0  = blockIdx.y * 16;
    const int f0   = blockIdx.x * 16;
    const int K    = ci * 12;
    const int Wp   = Fin + 2;
    const int arow = (co0 + l15 < co) ? (co0 + l15) : (co - 1);  // clamped, masked at store
    const int fcol = f0 + l15;
    const int fc   = (fcol < Fout) ? fcol : (Fout - 1);          // clamped, masked at store
    const int fcs  = fc * stride;     // padded base column (pad=1 folds into kf)

    // lane-dependent but loop-invariant tap offsets: off = kt*Wp + kf for the
    // two K values this lane holds at rr = 0,4,8.
    int off0[3], off1[3];
#pragma unroll
    for (int u = 0; u < 3; ++u) {
        int rd = 4 * u + koff;
        int kt0 = rd / 3, kf0 = rd - 3 * kt0;
        int kt1 = (rd + 1) / 3, kf1 = (rd + 1) - 3 * kt1;
        off0[u] = kt0 * Wp + kf0 + fcs;
        off1[u] = kt1 * Wp + kf1 + fcs;
    }

    v8f acc = {};
    const float* wrow = W + (size_t)arow * K + koff;
#pragma unroll 2
    for (int c = 0; c < ci; ++c) {
        const float* rb = inpP + (size_t)c * 4 * Wp;
        const float* wc = wrow + (size_t)c * 12;
#pragma unroll
        for (int u = 0; u < 3; ++u) {
            v2f a = *(const v2f*)(wc + 4 * u);   // b64, 8B-aligned
            v2f b;
            b.x = rb[off0[u]];
            b.y = rb[off1[u]];
            acc = __builtin_amdgcn_wmma_f32_16x16x4_f32(
                false, a, false, b, (short)0, acc, false, false);
        }
    }

    const int pWp = Fout + 2;
#pragma unroll
    for (int r = 0; r < 8; ++r) {
        int row = co0 + half * 8 + r;            // C/D layout: VGPR r -> M=r / r+8
        int col = f0 + l15;
        if (row < co && col < Fout) {
            float v = acc[r] + bias[row];
            if (flags & 1) {
                v = bng[row] * (v - bnm[row]) * rsqrtf(bnv[row] + 1e-5f) + bnb[row];
                v = elu1(v);
            }
            if (flags & 2) v += addp[(size_t)row * Fout + col];
            out[(size_t)row * Fout + col] = v;
            if (packp) packp[((size_t)row * 4 + 3) * pWp + 1 + col] = v;
        }
    }
}

// ------------------------- decoder skip 1x1 conv (WMMA) --------------------
// out = addp + W @ xen + bias     W:(ci,ci)  xen:(ci,F); also fills row 3 of
// the decoder res-conv pack (width F+2).
__global__ void k_skip1x1(const float* __restrict__ xen,
                          const float* __restrict__ W,
                          const float* __restrict__ bias,
                          const float* __restrict__ addp,
                          float* __restrict__ out,
                          float* __restrict__ packp,
                          int ci, int F) {
    const int lane = threadIdx.x;
    const int half = lane >> 4;
    const int l15  = lane & 15;
    const int koff = half * 2;
    const int co0  = blockIdx.y * 16;
    const int f0   = blockIdx.x * 16;
    const int arow = (co0 + l15 < ci) ? (co0 + l15) : (ci - 1);
    const int fcol = f0 + l15;
    const int fc   = (fcol < F) ? fcol : (F - 1);   // clamped, masked at store

    v8f acc = {};
    const float* wrow = W + (size_t)arow * ci + koff;
    for (int k = 0; k < ci; k += 4) {
        v2f a = *(const v2f*)(wrow + k);            // b64
        v2f b;
        b.x = xen[(size_t)(k + koff) * F + fc];
        b.y = xen[(size_t)(k + koff + 1) * F + fc];
        acc = __builtin_amdgcn_wmma_f32_16x16x4_f32(
            false, a, false, b, (short)0, acc, false, false);
    }
    const int pWp = F + 2;
#pragma unroll
    for (int r = 0; r < 8; ++r) {
        int row = co0 + half * 8 + r;
        int col = f0 + l15;
        if (row < ci && col < F) {
            float v = acc[r] + bias[row] + addp[(size_t)row * F + col];
            out[(size_t)row * F + col] = v;
            packp[((size_t)row * 4 + 3) * pWp + 1 + col] = v;
        }
    }
}

// ------------------------- matvec: out = W @ v + b -------------------------
__global__ void k_matvec(const float* __restrict__ W, const float* __restrict__ v,
                         const float* __restrict__ b, float* __restrict__ out,
                         int M, int K) {
    int wid  = blockIdx.x * (blockDim.x >> 5) + (threadIdx.x >> 5);
    int lane = threadIdx.x & 31;
    if (wid >= M) return;
    const float4* wr = (const float4*)(W + (size_t)wid * K);
    const float4* vv = (const float4*)v;
    float acc = 0.0f;
    int K4 = K >> 2;
    for (int k = lane; k < K4; k += 32) {
        float4 a = wr[k];
        float4 c = vv[k];
        acc += a.x * c.x + a.y * c.y + a.z * c.z + a.w * c.w;
    }
#pragma unroll
    for (int off = 16; off > 0; off >>= 1) acc += __shfl_xor(acc, off, 32);
    if (lane == 0) out[wid] = acc + b[wid];
}

// ------------------------- GRU gate pointwise ------------------------------
__global__ void k_gru(const float* __restrict__ gi, const float* __restrict__ gh,
                      const float* __restrict__ hp, float* __restrict__ hnew,
                      float* __restrict__ hout) {
    int t = blockIdx.x * blockDim.x + threadIdx.x;
    if (t >= 576) return;
    float r = sigm(gi[t] + gh[t]);
    float z = sigm(gi[576 + t] + gh[576 + t]);
    float n = tanhf(gi[1152 + t] + r * gh[1152 + t]);
    float h = (1.0f - z) * n + z * hp[t];
    hnew[t] = h;
    hout[t] = h;
}

// ------------------------- subpixel (freq x2) + BN + ELU + crop ------------
__global__ void k_subpix(const float* __restrict__ tmp,
                         const float* __restrict__ g, const float* __restrict__ b,
                         const float* __restrict__ m, const float* __restrict__ vv,
                         float* __restrict__ out, int co, int F, int Fc) {
    int idx = blockIdx.x * blockDim.x + threadIdx.x;
    if (idx >= co * Fc) return;
    int c = idx / Fc, f = idx - c * Fc;
    float v = (f < F) ? tmp[(size_t)c * F + f] : tmp[(size_t)(co + c) * F + (f - F)];
    v = g[c] * (v - m[c]) * rsqrtf(vv[c] + 1e-5f) + b[c];
    out[idx] = elu1(v);
}

// ------------------------- complex conv mask (CCM) -------------------------
__global__ void k_ccm(const float* __restrict__ d1, const float* __restrict__ x,
                      const float* __restrict__ mc, float* __restrict__ enh,
                      float* __restrict__ mco) {
    int f = blockIdx.x * blockDim.x + threadIdx.x;
    if (f >= 257) return;
    const float S3 = 0.86602540378443864676f;  // sqrt(3)/2
    float er = 0.0f, ei = 0.0f;
#pragma unroll
    for (int p = 0; p < 3; ++p) {
#pragma unroll
        for (int q = 0; q < 3; ++q) {
            int k = p * 3 + q;
            float m0 = d1[(size_t)k * 257 + f];
            float m1 = d1[(size_t)(9 + k) * 257 + f];
            float m2 = d1[(size_t)(18 + k) * 257 + f];
            float Hr = m0 - 0.5f * (m1 + m2);
            float Hi = S3 * (m1 - m2);
            int g = f + q - 1;                   // freq pad +/-1
            float xr = 0.0f, xi = 0.0f;
            if (g >= 0 && g < 257) {
                if (p < 2) { xr = mc[g * 4 + p * 2]; xi = mc[g * 4 + p * 2 + 1]; }
                else       { xr = x[g * 2];          xi = x[g * 2 + 1]; }
            }
            er += Hr * xr - Hi * xi;
            ei += Hr * xi + Hi * xr;
        }
    }
    enh[f * 2]     = er;
    enh[f * 2 + 1] = ei;
    mco[f * 4 + 0] = mc[f * 4 + 2];
    mco[f * 4 + 1] = mc[f * 4 + 3];
    mco[f * 4 + 2] = x[f * 2];
    mco[f * 4 + 3] = x[f * 2 + 1];
}

// ------------------------- batched cache updates ---------------------------
// out(C,3,F) = concat(cache_in[:,1:,:], x(C,F)) along time.
struct CacheJob { const float* cin; const float* x; float* out; int C; int F; };
struct CacheJobs { CacheJob j[20]; };

__global__ void k_caches(CacheJobs jobs) {
    CacheJob jb = jobs.j[blockIdx.y];
    int n = jb.C * 3 * jb.F;
    for (int e = blockIdx.x * blockDim.x + threadIdx.x; e < n;
         e += gridDim.x * blockDim.x) {
        int f = e % jb.F;
        int t = (e / jb.F) % 3;
        int c = e / (3 * jb.F);
        jb.out[e] = (t < 2) ? jb.cin[((size_t)c * 3 + t + 1) * jb.F + f]
                            : jb.x[(size_t)c * jb.F + f];
    }
}

// ---------------------------------------------------------------------------
// Input layout: top-level setup_inputs() insertion order (x, 22 caches,
// params). 'params' flattened jax-pytree style (sorted dict keys):
//   dec[j]: bn(g,beta,mean,var), db, dw, rb, rbn(4), rw, sb, sw   (14 each)
//   enc[i]: bn(4), cb, cw, rb, rbn(4), rw                         (12 each)
//   gru: b_hh, b_ih, fc_b, fc_w, w_hh, w_ih                       (6)
// ---------------------------------------------------------------------------
extern "C" void kernel_launch(void* const* d_in, const int* in_sizes, int n_in,
                              void* d_out, int out_size, void* d_ws, size_t ws_size,
                              hipStream_t stream) {
    auto F32 = [&](int i) { return (const float*)d_in[i]; };

    const float* x      = F32(0);
    const float* ec_in[5] = {F32(1), F32(3), F32(5), F32(7), F32(9)};
    const float* er_in[5] = {F32(2), F32(4), F32(6), F32(8), F32(10)};
    const float* h_in   = F32(11);
    const float* dc_in[5] = {F32(12), F32(14), F32(16), F32(18), F32(20)};
    const float* dr_in[5] = {F32(13), F32(15), F32(17), F32(19), F32(21)};
    const float* mc_in  = F32(22);

    const int DEC0 = 23;   // + j*14
    const int ENC0 = 93;   // + i*12
    const float* g_bhh = F32(153);
    const float* g_bih = F32(154);
    const float* g_fcb = F32(155);
    const float* g_fcw = F32(156);
    const float* g_whh = F32(157);
    const float* g_wih = F32(158);

    float* wsf = (float*)d_ws;
    size_t cur = 0;
    auto alloc = [&](size_t n) { float* p = wsf + cur; cur += (n + 3) & ~(size_t)3; return p; };

    const int ci_e[5]  = {2, 64, 128, 128, 128};
    const int co_e[5]  = {64, 128, 128, 128, 128};
    const int Fin_e[5] = {257, 129, 65, 33, 17};
    const int Fout_e[5]= {129, 65, 33, 17, 9};

    const int ci_d[5] = {128, 128, 128, 128, 64};
    const int co_d[5] = {128, 128, 128, 64, 27};
    const int F_d[5]  = {9, 17, 33, 65, 129};
    const int Fc_d[5] = {17, 33, 65, 129, 257};

    // plain activation buffers
    float* enbuf[6];
    enbuf[0] = alloc(2 * 257);
    float* ybuf[5];
    for (int i = 0; i < 5; ++i) {
        ybuf[i]      = alloc((size_t)co_e[i] * Fout_e[i]);
        enbuf[i + 1] = alloc((size_t)co_e[i] * Fout_e[i]);
    }
    float* gi    = alloc(1728);
    float* gh    = alloc(1728);
    float* hnew  = alloc(576);
    float* fcout = alloc(1152);
    float* yd[5]; float* y2[5]; float* tmpd[5]; float* dbuf[5];
    for (int j = 0; j < 5; ++j) {
        yd[j]   = alloc((size_t)ci_d[j] * F_d[j]);
        y2[j]   = alloc((size_t)ci_d[j] * F_d[j]);
        tmpd[j] = alloc((size_t)2 * co_d[j] * F_d[j]);
        dbuf[j] = alloc((size_t)co_d[j] * Fc_d[j]);
    }
    // packed conv inputs (C,4,F+2)
    float* pack_ec[5]; float* pack_er[5]; float* pack_dr[5]; float* pack_dc[5];
    for (int i = 0; i < 5; ++i) {
        pack_ec[i] = alloc((size_t)ci_e[i] * 4 * (Fin_e[i] + 2));
        pack_er[i] = alloc((size_t)co_e[i] * 4 * (Fout_e[i] + 2));
    }
    for (int j = 0; j < 5; ++j) {
        pack_dr[j] = alloc((size_t)ci_d[j] * 4 * (F_d[j] + 2));
        pack_dc[j] = alloc((size_t)ci_d[j] * 4 * (F_d[j] + 2));
    }

    // output offsets (return-tuple order)
    float* dof = (float*)d_out;
    size_t oc = 0;
    auto outp = [&](size_t n) { float* p = dof + oc; oc += n; return p; };
    float* o_enh = outp(514);
    float* o_ec[5]; float* o_er[5];
    o_ec[0] = outp(1542);  o_er[0] = outp(24768);
    o_ec[1] = outp(24768); o_er[1] = outp(24960);
    o_ec[2] = outp(24960); o_er[2] = outp(12672);
    o_ec[3] = outp(12672); o_er[3] = outp(6528);
    o_ec[4] = outp(6528);  o_er[4] = outp(3456);
    float* o_h = outp(576);
    float* o_dc[5]; float* o_dr[5];
    o_dc[0] = outp(3456);  o_dr[0] = outp(3456);
    o_dc[1] = outp(6528);  o_dr[1] = outp(6528);
    o_dc[2] = outp(12672); o_dr[2] = outp(12672);
    o_dc[3] = outp(24960); o_dr[3] = outp(24960);
    o_dc[4] = outp(24768); o_dr[4] = outp(24768);
    float* o_mc = outp(1028);

    // ---- 0. init all packed inputs (cache rows + zero pad + zero row 3) ----
    PackJobs pjobs;
    for (int i = 0; i < 5; ++i) {
        pjobs.j[i]     = {ec_in[i], pack_ec[i], ci_e[i], Fin_e[i]};
        pjobs.j[5 + i] = {er_in[i], pack_er[i], co_e[i], Fout_e[i]};
    }
    for (int j = 0; j < 5; ++j) {
        pjobs.j[10 + j] = {dr_in[j], pack_dr[j], ci_d[j], F_d[j]};
        pjobs.j[15 + j] = {dc_in[j], pack_dc[j], ci_d[j], F_d[j]};
    }
    k_pack_caches<<<dim3(136, 20), 256, 0, stream>>>(pjobs);

    // ---- 1. feature extraction (fills pack_ec[0] row 3) ----
    k_fe<<<dim3(5), dim3(64), 0, stream>>>(x, enbuf[0], pack_ec[0]);

    // ---- 2. encoder ----
    for (int i = 0; i < 5; ++i) {
        int base = ENC0 + i * 12;
        const float *bng = F32(base), *bnb = F32(base + 1),
                    *bnm = F32(base + 2), *bnv = F32(base + 3);
        const float *cb = F32(base + 4), *cw = F32(base + 5);
        const float *rb = F32(base + 6);
        const float *rg = F32(base + 7), *rbb = F32(base + 8),
                    *rm = F32(base + 9), *rv = F32(base + 10);
        const float *rw = F32(base + 11);
        dim3 grid((Fout_e[i] + 15) / 16, (co_e[i] + 15) / 16);
        k_conv<<<grid, 32, 0, stream>>>(pack_ec[i], cw, cb,
                                        bng, bnb, bnm, bnv, nullptr,
                                        ybuf[i], pack_er[i],
                                        ci_e[i], co_e[i], Fin_e[i], Fout_e[i], 2, 1);
        k_conv<<<grid, 32, 0, stream>>>(pack_er[i], rw, rb,
                                        rg, rbb, rm, rv, ybuf[i],
                                        enbuf[i + 1], (i < 4) ? pack_ec[i + 1] : nullptr,
                                        co_e[i], co_e[i], Fout_e[i], Fout_e[i], 1, 3);
    }

    // ---- 3. GRU ----
    k_matvec<<<dim3((1728 + 7) / 8), 256, 0, stream>>>(g_wih, enbuf[5], g_bih, gi, 1728, 1152);
    k_matvec<<<dim3((1728 + 7) / 8), 256, 0, stream>>>(g_whh, h_in, g_bhh, gh, 1728, 576);
    k_gru<<<dim3(3), 256, 0, stream>>>(gi, gh, h_in, hnew, o_h);
    k_matvec<<<dim3((1152 + 7) / 8), 256, 0, stream>>>(g_fcw, hnew, g_fcb, fcout, 1152, 576);

    // ---- 4. decoder ----
    const float* dprev = fcout;
    for (int j = 0; j < 5; ++j) {
        int base = DEC0 + j * 14;
        const float *bng = F32(base), *bnb = F32(base + 1),
                    *bnm = F32(base + 2), *bnv = F32(base + 3);
        const float *db = F32(base + 4), *dw = F32(base + 5);
        const float *rb = F32(base + 6);
        const float *rg = F32(base + 7), *rbb = F32(base + 8),
                    *rm = F32(base + 9), *rv = F32(base + 10);
        const float *rw = F32(base + 11);
        const float *sb = F32(base + 12), *sw = F32(base + 13);
        const float* skip = enbuf[5 - j];
        int ci = ci_d[j], co = co_d[j], F = F_d[j], Fc = Fc_d[j];

        dim3 gs((F + 15) / 16, (ci + 15) / 16);
        k_skip1x1<<<gs, 32, 0, stream>>>(skip, sw, sb, dprev, yd[j], pack_dr[j], ci, F);
        k_conv<<<gs, 32, 0, stream>>>(pack_dr[j], rw, rb,
                                      rg, rbb, rm, rv, yd[j],
                                      y2[j], pack_dc[j],
                                      ci, ci, F, F, 1, 3);
        dim3 gd((F + 15) / 16, (2 * co + 15) / 16);
        k_conv<<<gd, 32, 0, stream>>>(pack_dc[j], dw, db,
                                      nullptr, nullptr, nullptr, nullptr, nullptr,
                                      tmpd[j], nullptr,
                                      ci, 2 * co, F, F, 1, 0);
        k_subpix<<<dim3((co * Fc + 255) / 256), 256, 0, stream>>>(
            tmpd[j], bng, bnb, bnm, bnv, dbuf[j], co, F, Fc);
        dprev = dbuf[j];
    }

    // ---- 5. complex conv mask + m_cache update ----
    k_ccm<<<dim3(5), 64, 0, stream>>>(dbuf[4], x, mc_in, o_enh, o_mc);

    // ---- 6. all 20 conv/res cache updates in one launch ----
    CacheJobs jobs;
    for (int i = 0; i < 5; ++i) {
        jobs.j[i]     = {ec_in[i], enbuf[i], o_ec[i], ci_e[i], Fin_e[i]};
        jobs.j[5 + i] = {er_in[i], ybuf[i],  o_er[i], co_e[i], Fout_e[i]};
    }
    for (int j = 0; j < 5; ++j) {
        jobs.j[10 + j] = {dr_in[j], yd[j], o_dr[j], ci_d[j], F_d[j]};
        jobs.j[15 + j] = {dc_in[j], y2[j], o_dc[j], ci_d[j], F_d[j]};
    }
    k_caches<<<dim3(98, 20), 256, 0, stream>>>(jobs);
}